// SpatialTransformer_90391881712155
// MI455X (gfx1250) — compile-verified
//
#include <hip/hip_runtime.h>

// Problem constants from the reference: x (2,2,160,192,160) f32, flow (2,3,160,192,160) f32
#define DD 160
#define HH 192
#define WW 160
#define DHW (DD * HH * WW)          // 4,915,200
#define BB 2
#define CPLANE (DHW)                // one channel plane
#define PREF_AHEAD 2048             // elements (~8KB) lookahead for streaming prefetch

__device__ __forceinline__ int iclamp(int v, int lo, int hi) {
    return v < lo ? lo : (v > hi ? hi : v);
}

__global__ __launch_bounds__(256) void warp3d_trilinear_kernel(
    const float* __restrict__ x,
    const float* __restrict__ flow,
    float* __restrict__ out)
{
    const int idx = blockIdx.x * 256 + threadIdx.x;   // [0, BB*DHW)
    const int b = (idx >= DHW) ? 1 : 0;
    const int s = idx - b * DHW;                      // spatial index within batch

    // decompose s -> (d, h, w)
    const int w = s % WW;
    const int t = s / WW;
    const int h = t % HH;
    const int d = t / HH;

    // flow planes for this batch: [0]=dz, [1]=dy, [2]=dx
    const float* __restrict__ fb = flow + (size_t)b * 3 * DHW;

    // CDNA5 path: global_prefetch_b8 for the streaming flow planes and the
    // identity-neighborhood of x (samples cluster around grid + N(0,1)).
    if (s + PREF_AHEAD < DHW) {
        __builtin_prefetch(fb + s + PREF_AHEAD, 0, 1);
        __builtin_prefetch(fb + CPLANE + s + PREF_AHEAD, 0, 1);
        __builtin_prefetch(fb + 2 * CPLANE + s + PREF_AHEAD, 0, 1);
        __builtin_prefetch(x + ((size_t)b * 2 + 0) * DHW + s + PREF_AHEAD, 0, 1);
        __builtin_prefetch(x + ((size_t)b * 2 + 1) * DHW + s + PREF_AHEAD, 0, 1);
    }

    // sample position = grid + flow (voxel coords); normalization cancels (see reference)
    const float pz = (float)d + fb[s];
    const float py = (float)h + fb[CPLANE + s];
    const float px = (float)w + fb[2 * CPLANE + s];

    const float z0f = floorf(pz);
    const float y0f = floorf(py);
    const float x0f = floorf(px);
    const float fz = pz - z0f;
    const float fy = py - y0f;
    const float fx = px - x0f;
    const int z0 = (int)z0f;
    const int y0 = (int)y0f;
    const int x0 = (int)x0f;

    const float wz[2] = {1.0f - fz, fz};
    const float wy[2] = {1.0f - fy, fy};
    const float wx[2] = {1.0f - fx, fx};

    const float* __restrict__ xc0 = x + ((size_t)b * 2 + 0) * DHW;  // channel 0 plane
    const float* __restrict__ xc1 = xc0 + DHW;                      // channel 1 plane

    float acc0 = 0.0f;
    float acc1 = 0.0f;

#pragma unroll
    for (int dz = 0; dz < 2; ++dz) {
        const int zi = z0 + dz;
        const bool inz = (unsigned)zi < (unsigned)DD;
        const int zc = iclamp(zi, 0, DD - 1);
        const float wzv = wz[dz];
#pragma unroll
        for (int dy = 0; dy < 2; ++dy) {
            const int yi = y0 + dy;
            const bool iny = (unsigned)yi < (unsigned)HH;
            const int yc = iclamp(yi, 0, HH - 1);
            const int rowbase = (zc * HH + yc) * WW;
            const float wzy = wzv * wy[dy];
#pragma unroll
            for (int dx = 0; dx < 2; ++dx) {
                const int xi = x0 + dx;
                const bool inx = (unsigned)xi < (unsigned)WW;
                const int xc = iclamp(xi, 0, WW - 1);
                // zeros padding: clamp the gather address, zero the weight if OOB
                const float wgt = wzy * wx[dx] * (float)(inz & iny & inx);
                const int off = rowbase + xc;
                acc0 = fmaf(xc0[off], wgt, acc0);
                acc1 = fmaf(xc1[off], wgt, acc1);
            }
        }
    }

    out[((size_t)b * 2 + 0) * DHW + s] = acc0;
    out[((size_t)b * 2 + 1) * DHW + s] = acc1;
}

extern "C" void kernel_launch(void* const* d_in, const int* in_sizes, int n_in,
                              void* d_out, int out_size, void* d_ws, size_t ws_size,
                              hipStream_t stream) {
    (void)in_sizes; (void)n_in; (void)out_size; (void)d_ws; (void)ws_size;
    const float* x    = (const float*)d_in[0];   // (2,2,160,192,160) f32
    const float* flow = (const float*)d_in[1];   // (2,3,160,192,160) f32
    float* out        = (float*)d_out;           // (2,2,160,192,160) f32

    const int total  = BB * DHW;                 // 9,830,400 threads (1 per voxel, 2 channels each)
    const int blocks = total / 256;              // exact: 38,400
    warp3d_trilinear_kernel<<<blocks, 256, 0, stream>>>(x, flow, out);
}